// HashEncodingPyTorch_87436944212735
// MI455X (gfx1250) — compile-verified
//
#include <hip/hip_runtime.h>
#include <cmath>

// -----------------------------------------------------------------------------
// Multiresolution hash-grid encoding (Instant-NGP style) for MI455X / gfx1250.
//
// Memory-bound gather kernel. The entire embedding table (64 MB) fits in the
// 192 MB L2, so gathers use default RT caching; the 256 MB streaming output
// and the 24 MB streaming input use CDNA5 non-temporal hints (TH=NT) so they
// do not evict the table from L2. WMMA is deliberately NOT used: trilinear
// interpolation is a per-lane 8-element dot product with per-lane operands,
// which the wave-shared WMMA operand model cannot express.
// -----------------------------------------------------------------------------

#define HLVL 16
#define HTBL 524288u          // 2^19 entries per level
#define HMASK 0x7FFFFu        // % T == & (T-1)
#define HP1 2654435761u
#define HP2 805459861u

typedef float v4f __attribute__((ext_vector_type(4)));

struct ResParams { float res[HLVL]; };

__global__ __launch_bounds__(256) void hash_encode_kernel(
    const float* __restrict__ x,      // (N, 3) f32
    const float* __restrict__ emb,    // (16, 524288, 2) f32
    float* __restrict__ out,          // (N, 32) f32
    int N, ResParams rp)
{
    const int p = blockIdx.x * 256 + threadIdx.x;
    if (p >= N) return;

    // Streaming, read-once input: non-temporal loads (do not pollute L2).
    const size_t xb = 3ull * (size_t)p;
    const float px = __builtin_nontemporal_load(x + xb + 0);
    const float py = __builtin_nontemporal_load(x + xb + 1);
    const float pz = __builtin_nontemporal_load(x + xb + 2);

    float fout[2 * HLVL];
    const float2* __restrict__ tbl0 = (const float2*)emb;

#pragma unroll
    for (int i = 0; i < HLVL; ++i) {
        const float r  = rp.res[i];            // compile-time index: stays in SGPR
        const float xs = px * r;
        const float ys = py * r;
        const float zs = pz * r;
        // x in [0,1), r <= 2048 -> truncation toward zero == floor, fits u32.
        const int xi = (int)xs, yi = (int)ys, zi = (int)zs;
        const float fx = xs - (float)xi;
        const float fy = ys - (float)yi;
        const float fz = zs - (float)zi;

        // Hash components; low 19 bits identical to the int64 reference.
        const unsigned ax0 = (unsigned)xi;            // * PRIMES[0] == 1
        const unsigned ax1 = ax0 + 1u;
        const unsigned by0 = (unsigned)yi * HP1;
        const unsigned by1 = by0 + HP1;
        const unsigned cz0 = (unsigned)zi * HP2;
        const unsigned cz1 = cz0 + HP2;

        const float2* __restrict__ t = tbl0 + (size_t)i * HTBL;

        // Corner (dx,dy,dz); gathers are 8B global_load_b64, L2-resident table.
        const float2 e000 = t[(ax0 ^ by0 ^ cz0) & HMASK];
        const float2 e001 = t[(ax0 ^ by0 ^ cz1) & HMASK];
        const float2 e010 = t[(ax0 ^ by1 ^ cz0) & HMASK];
        const float2 e011 = t[(ax0 ^ by1 ^ cz1) & HMASK];
        const float2 e100 = t[(ax1 ^ by0 ^ cz0) & HMASK];
        const float2 e101 = t[(ax1 ^ by0 ^ cz1) & HMASK];
        const float2 e110 = t[(ax1 ^ by1 ^ cz0) & HMASK];
        const float2 e111 = t[(ax1 ^ by1 ^ cz1) & HMASK];

        const float gx = 1.0f - fx, gy = 1.0f - fy, gz = 1.0f - fz;

        // Same lerp structure as the reference (x, then y, then z).
        const float c00a = e000.x * gx + e100.x * fx;
        const float c00b = e000.y * gx + e100.y * fx;
        const float c01a = e001.x * gx + e101.x * fx;
        const float c01b = e001.y * gx + e101.y * fx;
        const float c10a = e010.x * gx + e110.x * fx;
        const float c10b = e010.y * gx + e110.y * fx;
        const float c11a = e011.x * gx + e111.x * fx;
        const float c11b = e011.y * gx + e111.y * fx;

        const float c0a = c00a * gy + c10a * fy;
        const float c0b = c00b * gy + c10b * fy;
        const float c1a = c01a * gy + c11a * fy;
        const float c1b = c01b * gy + c11b * fy;

        fout[2 * i + 0] = c0a * gz + c1a * fz;
        fout[2 * i + 1] = c0b * gz + c1b * fz;
    }

    // 128 B per point, 128 B aligned. 8 back-to-back NT b128 stores jointly
    // fill whole cachelines; TH=NT keeps the 256 MB output stream out of the
    // L2 so the 64 MB embedding table stays resident.
    v4f* __restrict__ op = (v4f*)(out + 32ull * (size_t)p);
#pragma unroll
    for (int q = 0; q < 8; ++q) {
        v4f v;
        v.x = fout[4 * q + 0];
        v.y = fout[4 * q + 1];
        v.z = fout[4 * q + 2];
        v.w = fout[4 * q + 3];
        __builtin_nontemporal_store(v, op + q);
    }
}

extern "C" void kernel_launch(void* const* d_in, const int* in_sizes, int n_in,
                              void* d_out, int out_size, void* d_ws, size_t ws_size,
                              hipStream_t stream) {
    const float* x   = (const float*)d_in[0];   // (N, 3) float32
    const float* emb = (const float*)d_in[1];   // (16, 524288, 2) float32
    float* out = (float*)d_out;                 // (N, 32) float32

    const int N = in_sizes[0] / 3;

    // Replicate Python's RESOLUTIONS computation with the *same* sequence of
    // double-precision libm ops so the floor() boundary cases (esp. i=15,
    // where 16*b^15 == 2048 +/- ~1e-11) resolve identically:
    //   _b = exp((log(2048) - log(16)) / 15);  res_i = floor(16 * _b**i)
    ResParams rp;
    const double b = exp((log(2048.0) - log(16.0)) / 15.0);
    for (int i = 0; i < HLVL; ++i) {
        rp.res[i] = (float)floor(16.0 * pow(b, (double)i));
    }

    const int blocks = (N + 255) / 256;
    hipLaunchKernelGGL(hash_encode_kernel, dim3(blocks), dim3(256), 0, stream,
                       x, emb, out, N, rp);
}